// attention_kivi_77506979824107
// MI455X (gfx1250) — compile-verified
//
#include <hip/hip_runtime.h>

// ---------------------------------------------------------------------------
// KIVI-style quantized-KV decode attention for MI455X (gfx1250).
//
// Roofline: ~520 MB of int32 4-bit codes must stream once -> ~22us at
// 23.3 TB/s; compute is ~264 MFLOP (negligible). Design: read codes once,
// coalesced; fold the per-group affine dequant into precomputed per-(b,h)
// operands (qs/bias for K, ps/bias2 for V) so the streaming passes are pure
// code-matmuls, executed with V_WMMA_F32_16X16X4_F32 accumulator chains.
// ---------------------------------------------------------------------------

typedef float v2f __attribute__((ext_vector_type(2)));
typedef float v8f __attribute__((ext_vector_type(8)));

#define TQ    3968
#define DIMD  128
#define NG    62            // TQ / 64 groups along kv dim (K cache)
#define NFULL 65            // 64 residual + 1 new
#define NT    (TQ + NFULL)  // 4033 total scores
#define QK_SCALE 0.088388347648318447f  // 1/sqrt(128)
#define BS  512
#define NW  (BS / 32)

__global__ __launch_bounds__(BS) void kivi_decode_kernel(
    const float* __restrict__ qg,  const float* __restrict__ kng,
    const float* __restrict__ vng, const float* __restrict__ kfg,
    const float* __restrict__ vfg, const float* __restrict__ kst,
    const float* __restrict__ kmt, const float* __restrict__ vsc,
    const float* __restrict__ vmn, const int* __restrict__ kq,
    const int* __restrict__ vq,    float* __restrict__ outg)
{
    const int bh   = blockIdx.x;          // b*32 + h
    const int tid  = threadIdx.x;
    const int lane = tid & 31;
    const int wave = tid >> 5;

    extern __shared__ float sm[];
    float* qs    = sm;               // [NG][130] padded; later aliased as ps[2][TQ]
    float* sc    = qs + NG * 130;    // 4096 (scores -> probs)
    float* q_l   = sc + 4096;        // 128
    float* biasg = q_l + DIMD;       // 64
    float* oacc  = biasg + 64;       // 128
    float* zbuf  = oacc + DIMD;      // 128 (zero source for WMMA A rows 1..15)
    float* red   = zbuf + DIMD;      // 64  (block-reduction scratch)

    const size_t bhD = (size_t)bh * DIMD;
    const float* qp  = qg  + bhD;
    const float* knp = kng + bhD;
    const float* vnp = vng + bhD;
    const float* kfp = kfg + (size_t)bh * 64 * DIMD;
    const float* vfp = vfg + (size_t)bh * 64 * DIMD;
    const float* ksp = kst + (size_t)bh * DIMD * NG;
    const float* kmp = kmt + (size_t)bh * DIMD * NG;
    const float* vsp = vsc + (size_t)bh * TQ * 2;
    const float* vmp = vmn + (size_t)bh * TQ * 2;
    const int*   kqp = kq  + (size_t)bh * DIMD * TQ;   // [d][t]
    const int*   vqp = vq  + (size_t)bh * TQ * DIMD;   // [t][d]

    // ---- Phase A: init -----------------------------------------------------
    for (int i = tid; i < DIMD; i += BS) { q_l[i] = qp[i]; oacc[i] = 0.f; zbuf[i] = 0.f; }
    for (int i = tid; i < 64;   i += BS) biasg[i] = 0.f;
    __syncthreads();

    // ---- Phase B: qs[g][d] = q[d]*s[d][g];  biasg[g] = sum_d q[d]*m[d][g] --
    for (int i = tid; i < DIMD * NG; i += BS) {
        const int d = i / NG;
        const int g = i - d * NG;
        const float qd = q_l[d];
        qs[g * 130 + d] = qd * ksp[i];
        atomicAdd(&biasg[g], qd * kmp[i]);
    }
    __syncthreads();

    const bool alane   = (lane == 0) || (lane == 16);
    const int  halfoff = (lane >> 4) << 1;       // 0 (K=0,1) or 2 (K=2,3)

    // ---- Phase C: quantized QK^T via WMMA f32 16x16x4 ----------------------
    // Tile = 16 t-columns; accumulate over d in 32 chunks of K=4.
    // A: row0 = qs chunk (lanes 0/16 only); B: codes, coalesced per 16-lane half.
    for (int tile = wave; tile < TQ / 16; tile += NW) {
        const int t0 = tile << 4;
        const int g  = t0 >> 6;                   // group constant across tile
        const float* qsg = alane ? (qs + g * 130 + halfoff) : zbuf;
        const int* kcol  = kqp + (size_t)halfoff * TQ + t0 + (lane & 15);
        v8f acc = {};
        #pragma unroll 4
        for (int c4 = 0; c4 < 32; ++c4) {
            const int d0 = c4 << 2;
            v2f a;  a.x = qsg[d0];  a.y = qsg[d0 + 1];
            const int* p0 = kcol + (size_t)d0 * TQ;
            v2f bb; bb.x = (float)p0[0]; bb.y = (float)p0[TQ];
            acc = __builtin_amdgcn_wmma_f32_16x16x4_f32(
                false, a, false, bb, (short)0, acc, false, false);
        }
        if (lane < 16) sc[t0 + lane] = (acc[0] + biasg[g]) * QK_SCALE;
    }

    // ---- Phase D: full-precision keys (64 residual + 1 new) ----------------
    for (int r = wave; r < NFULL; r += NW) {
        const float* krow = (r < 64) ? (kfp + r * DIMD) : knp;
        const float4 kv = *(const float4*)(krow + lane * 4);
        float a = q_l[lane*4+0]*kv.x + q_l[lane*4+1]*kv.y
                + q_l[lane*4+2]*kv.z + q_l[lane*4+3]*kv.w;
        #pragma unroll
        for (int m = 16; m >= 1; m >>= 1) a += __shfl_xor(a, m, 32);
        if (lane == 0) sc[TQ + r] = a * QK_SCALE;
    }
    __syncthreads();

    // ---- Phase E: softmax over 4033 scores (in LDS) ------------------------
    float mx = -1e30f;
    for (int i = tid; i < NT; i += BS) mx = fmaxf(mx, sc[i]);
    #pragma unroll
    for (int m = 16; m >= 1; m >>= 1) mx = fmaxf(mx, __shfl_xor(mx, m, 32));
    if (lane == 0) red[wave] = mx;
    __syncthreads();
    if (tid == 0) {
        float v = red[0];
        for (int w = 1; w < NW; ++w) v = fmaxf(v, red[w]);
        red[48] = v;
    }
    __syncthreads();
    mx = red[48];

    // exp + build ps[g2][t] = p*v_scale and bias2 partials (ps aliases qs).
    float* ps = qs;                 // qs is dead; 2*TQ <= NG*130
    float se = 0.f, b20 = 0.f, b21 = 0.f;
    for (int i = tid; i < NT; i += BS) {
        const float e = __expf(sc[i] - mx);
        sc[i] = e;
        se += e;
        if (i < TQ) {
            const float2 s2 = *(const float2*)(vsp + i * 2);
            const float2 m2 = *(const float2*)(vmp + i * 2);
            ps[i]      = e * s2.x;
            ps[TQ + i] = e * s2.y;
            b20 += e * m2.x;
            b21 += e * m2.y;
        }
    }
    #pragma unroll
    for (int m = 16; m >= 1; m >>= 1) {
        se  += __shfl_xor(se,  m, 32);
        b20 += __shfl_xor(b20, m, 32);
        b21 += __shfl_xor(b21, m, 32);
    }
    if (lane == 0) { red[wave] = se; red[16 + wave] = b20; red[32 + wave] = b21; }
    __syncthreads();
    if (tid == 0) {
        float s = 0.f, x = 0.f, y = 0.f;
        for (int w = 0; w < NW; ++w) { s += red[w]; x += red[16 + w]; y += red[32 + w]; }
        red[49] = 1.0f / s;  red[50] = x;  red[51] = y;
    }
    __syncthreads();
    const float zinv  = red[49];
    const float b2lo  = red[50];
    const float b2hi  = red[51];

    // ---- Phase F: quantized PV via WMMA f32 16x16x4 ------------------------
    // Wave pair per 16-wide d-tile; each wave covers half the t range.
    {
        const int d0    = (wave >> 1) << 4;
        const int gd    = d0 >> 6;                 // 0 or 1 (d-group)
        const int tbase = (wave & 1) * (TQ / 2);
        const float* psg = ps + (size_t)gd * TQ;
        v8f acc = {};
        #pragma unroll 2
        for (int c4 = 0; c4 < TQ / 8; ++c4) {      // 496 chunks of K=4 t's
            const int t0 = tbase + (c4 << 2);
            const float* pp = alane ? (psg + t0 + halfoff) : zbuf;
            v2f a;  a.x = pp[0];  a.y = pp[1];
            const int* vp = vqp + (size_t)(t0 + halfoff) * DIMD + d0 + (lane & 15);
            v2f bb; bb.x = (float)vp[0]; bb.y = (float)vp[DIMD];
            acc = __builtin_amdgcn_wmma_f32_16x16x4_f32(
                false, a, false, bb, (short)0, acc, false, false);
        }
        if (lane < 16) atomicAdd(&oacc[d0 + lane], acc[0]);
    }

    // ---- Phase G: full-precision values ------------------------------------
    {
        const int dloc = tid & 127;
        const int part = tid >> 7;                 // 0..3
        float a = 0.f;
        for (int r = part; r < NFULL; r += 4) {
            const float* vrow = (r < 64) ? (vfp + r * DIMD) : vnp;
            a += sc[TQ + r] * vrow[dloc];
        }
        atomicAdd(&oacc[dloc], a);
    }
    __syncthreads();

    // ---- Phase H: write output ---------------------------------------------
    if (tid < DIMD) {
        const float b2 = (tid < 64) ? b2lo : b2hi;
        outg[bhD + tid] = (oacc[tid] + b2) * zinv;
    }
}

extern "C" void kernel_launch(void* const* d_in, const int* in_sizes, int n_in,
                              void* d_out, int out_size, void* d_ws, size_t ws_size,
                              hipStream_t stream) {
    (void)in_sizes; (void)n_in; (void)d_ws; (void)ws_size; (void)out_size;
    const float* q        = (const float*)d_in[0];
    const float* k        = (const float*)d_in[1];
    const float* v        = (const float*)d_in[2];
    const float* k_full   = (const float*)d_in[3];
    const float* v_full   = (const float*)d_in[4];
    const float* k_scale_t= (const float*)d_in[5];
    const float* k_mn_t   = (const float*)d_in[6];
    const float* v_scale  = (const float*)d_in[7];
    const float* v_mn     = (const float*)d_in[8];
    const int*   k_quant_t= (const int*)d_in[9];
    const int*   v_quant  = (const int*)d_in[10];
    float* out = (float*)d_out;

    const size_t smem = (size_t)(NG * 130 + 4096 + DIMD + 64 + DIMD + DIMD + 64)
                        * sizeof(float);   // 50672 B
    kivi_decode_kernel<<<dim3(128), dim3(BS), smem, stream>>>(
        q, k, v, k_full, v_full, k_scale_t, k_mn_t, v_scale, v_mn,
        k_quant_t, v_quant, out);
}